// GATBlock_26233660244458
// MI455X (gfx1250) — compile-verified
//
#include <hip/hip_runtime.h>
#include <cstdint>

typedef __attribute__((ext_vector_type(2))) float v2f;
typedef __attribute__((ext_vector_type(8))) float v8f;

// ---------------- helpers ----------------

__device__ __forceinline__ unsigned mapf(float f) {
    unsigned b = __float_as_uint(f);
    return (b & 0x80000000u) ? ~b : (b | 0x80000000u);
}
__device__ __forceinline__ float unmapf(unsigned u) {
    return __uint_as_float((u & 0x80000000u) ? (u ^ 0x80000000u) : ~u);
}
__device__ __forceinline__ float lrelu(float v, float s) {
    return (v > 0.f) ? v : s * v;
}

// ---------------- utility kernels ----------------

__global__ void zero_f32(float* __restrict__ p, long n, float v) {
    long i = (long)blockIdx.x * blockDim.x + threadIdx.x;
    long stride = (long)gridDim.x * blockDim.x;
    for (; i < n; i += stride) p[i] = v;
}

__global__ void zero_u32(unsigned* __restrict__ p, long n, unsigned v) {
    long i = (long)blockIdx.x * blockDim.x + threadIdx.x;
    long stride = (long)gridDim.x * blockDim.x;
    for (; i < n; i += stride) p[i] = v;
}

// column sums / sum-of-squares over [nrows x 128]; acc[0..127]=sum acc[128..255]=sumsq
__global__ void col_stats(const float* __restrict__ X, float* __restrict__ acc, int nrows) {
    const int c = threadIdx.x;  // 0..127
    float s = 0.f, s2 = 0.f;
    for (int r = blockIdx.x; r < nrows; r += gridDim.x) {
        float v = X[(size_t)r * 128 + c];
        s += v;
        s2 = fmaf(v, v, s2);
    }
    atomicAdd(&acc[c], s);
    atomicAdd(&acc[128 + c], s2);
}

// ab[c] = g*rstd ; ab[128+c] = b - mu*g*rstd  so bn(x) = x*a + c
__global__ void bn_finalize(const float* __restrict__ acc, const float* __restrict__ g,
                            const float* __restrict__ b, float* __restrict__ ab, int nrows) {
    const int c = threadIdx.x;  // 128 threads
    const float inv = 1.0f / (float)nrows;
    float mu = acc[c] * inv;
    float var = acc[128 + c] * inv - mu * mu;
    float rstd = rsqrtf(var + 1e-5f);
    float a = g[c] * rstd;
    ab[c] = a;
    ab[128 + c] = b[c] - mu * a;
}

// y = lrelu(x*a + c, 0.1)
__global__ void bn_apply_lrelu(const float* __restrict__ X, const float* __restrict__ ab,
                               float* __restrict__ Y, long total) {
    long i = (long)blockIdx.x * blockDim.x + threadIdx.x;
    if (i >= total) return;
    int c = (int)(i & 127);
    Y[i] = lrelu(fmaf(X[i], ab[c], ab[128 + c]), 0.1f);
}

// out = lrelu(acc + bias[col], 0.1)   (out may alias acc)
__global__ void add_bias_lrelu(const float* __restrict__ A, const float* __restrict__ bias,
                               float* __restrict__ out, long total) {
    long i = (long)blockIdx.x * blockDim.x + threadIdx.x;
    if (i >= total) return;
    out[i] = lrelu(A[i] + bias[i & 127], 0.1f);
}

// ---------------- dual-output WMMA GEMM ----------------
// XL = H @ Wl + bl ; XR = H @ Wr + br    (H: nrows x 128, W: 128 x 128 row-major)
// grid: (nrows/16, 8), block: 32 (one wave per 16x16 output tile)
__global__ void gemm_dual_wmma(const float* __restrict__ A,
                               const float* __restrict__ Wl, const float* __restrict__ bl,
                               const float* __restrict__ Wr, const float* __restrict__ br,
                               float* __restrict__ XL, float* __restrict__ XR) {
    __shared__ __align__(16) float sA[16 * 128];
    __shared__ __align__(16) float sBl[128 * 16];
    __shared__ __align__(16) float sBr[128 * 16];

    const int lane = threadIdx.x;  // 0..31
    const int row0 = blockIdx.x * 16;
    const int col0 = blockIdx.y * 16;

    // stage A tile (16x128 contiguous)
    {
        const float4* Ag = reinterpret_cast<const float4*>(A + (size_t)row0 * 128);
        float4* As = reinterpret_cast<float4*>(sA);
        for (int i = lane; i < 512; i += 32) As[i] = Ag[i];
    }
    // stage B tiles (128 rows x 16 cols)
    for (int i = lane; i < 512; i += 32) {
        int k = i >> 2, j = i & 3;
        const float4* wl = reinterpret_cast<const float4*>(Wl + (size_t)k * 128 + col0);
        const float4* wr = reinterpret_cast<const float4*>(Wr + (size_t)k * 128 + col0);
        reinterpret_cast<float4*>(sBl + k * 16)[j] = wl[j];
        reinterpret_cast<float4*>(sBr + k * 16)[j] = wr[j];
    }
    __syncthreads();

    // V_WMMA_F32_16X16X4_F32 fragment layout (wave32):
    //   A 16x4:  lane l -> m = l&15 ; vgpr j holds K = 2*(l>>4) + j
    //   B 4x16:  lane l -> n = l&15 ; vgpr j holds K = 2*(l>>4) + j
    //   C/D 16x16: lane l -> n = l&15 ; vgpr r holds M = r + 8*(l>>4)
    const int mn = lane & 15;
    const int kk = (lane >> 4) << 1;

    v8f accl = {};
    v8f accr = {};
#pragma unroll 4
    for (int kb = 0; kb < 128; kb += 4) {
        v2f a, b0, b1;
        a[0] = sA[mn * 128 + kb + kk];
        a[1] = sA[mn * 128 + kb + kk + 1];
        b0[0] = sBl[(kb + kk) * 16 + mn];
        b0[1] = sBl[(kb + kk + 1) * 16 + mn];
        b1[0] = sBr[(kb + kk) * 16 + mn];
        b1[1] = sBr[(kb + kk + 1) * 16 + mn];
        accl = __builtin_amdgcn_wmma_f32_16x16x4_f32(false, a, false, b0, (short)0, accl,
                                                     false, false);
        accr = __builtin_amdgcn_wmma_f32_16x16x4_f32(false, a, false, b1, (short)0, accr,
                                                     false, false);
    }

    const int half = lane >> 4;
    const float addl = bl[col0 + mn];
    const float addr_ = br[col0 + mn];
#pragma unroll
    for (int r = 0; r < 8; ++r) {
        size_t off = (size_t)(row0 + r + 8 * half) * 128 + col0 + mn;
        XL[off] = accl[r] + addl;
        XR[off] = accr[r] + addr_;
    }
}

// ---------------- edge kernels ----------------
// thread = (edge, head). total = 4*E.

// logits[e,h] = sum_c lrelu(xl[src] + xr[dst] + (ea@We), 0.2) * att[h,c]
// + ordered-uint atomicMax per (dst, head)
__global__ void edge_logits_max(const float* __restrict__ XL, const float* __restrict__ XR,
                                const float* __restrict__ EA, const int* __restrict__ src,
                                const int* __restrict__ dst, const float* __restrict__ We,
                                const float* __restrict__ att, float* __restrict__ logits,
                                unsigned* __restrict__ maxb, int total) {
    __shared__ float sWe[16 * 128];
    __shared__ float sAtt[128];
    for (int i = threadIdx.x; i < 16 * 128; i += blockDim.x) sWe[i] = We[i];
    if (threadIdx.x < 128) sAtt[threadIdx.x] = att[threadIdx.x];
    __syncthreads();

    int idx = blockIdx.x * blockDim.x + threadIdx.x;
    if (idx >= total) return;
    const int e = idx >> 2;
    const int h = idx & 3;
    const int s = src[e];
    const int d = dst[e];
    const float* xl = XL + (size_t)s * 128 + h * 32;
    const float* xr = XR + (size_t)d * 128 + h * 32;
    const float* eap = EA + (size_t)e * 16;

    float eav[16];
#pragma unroll
    for (int k = 0; k < 16; ++k) eav[k] = eap[k];

    float logit = 0.f;
#pragma unroll 8
    for (int c = 0; c < 32; ++c) {
        float eev = 0.f;
#pragma unroll
        for (int k = 0; k < 16; ++k) eev = fmaf(eav[k], sWe[k * 128 + h * 32 + c], eev);
        float v = xl[c] + xr[c] + eev;
        v = (v > 0.f) ? v : 0.2f * v;  // GATv2 internal slope 0.2
        logit = fmaf(v, sAtt[h * 32 + c], logit);
    }
    logits[idx] = logit;
    atomicMax(maxb + (size_t)d * 4 + h, mapf(logit));
}

// p = exp(logit - max[dst,h]) in place; ssum[dst,h] += p
__global__ void edge_softmax_sum(float* __restrict__ logits, const unsigned* __restrict__ maxb,
                                 const int* __restrict__ dst, float* __restrict__ ssum,
                                 int total) {
    int idx = blockIdx.x * blockDim.x + threadIdx.x;
    if (idx >= total) return;
    const int e = idx >> 2;
    const int h = idx & 3;
    const int d = dst[e];
    float mx = unmapf(maxb[(size_t)d * 4 + h]);
    float p = __expf(logits[idx] - mx);
    logits[idx] = p;
    atomicAdd(ssum + (size_t)d * 4 + h, p);
}

// Y[dst, h*32+c] += xl[src, h*32+c] * (p / ssum[dst,h])
__global__ void edge_aggregate(const float* __restrict__ p, const float* __restrict__ ssum,
                               const float* __restrict__ XL, const int* __restrict__ src,
                               const int* __restrict__ dst, float* __restrict__ Y, int total) {
    int idx = blockIdx.x * blockDim.x + threadIdx.x;
    if (idx >= total) return;
    const int e = idx >> 2;
    const int h = idx & 3;
    const int s = src[e];
    const int d = dst[e];
    float alpha = p[idx] / ssum[(size_t)d * 4 + h];
    const float* xl = XL + (size_t)s * 128 + h * 32;
    float* y = Y + (size_t)d * 128 + h * 32;
#pragma unroll 8
    for (int c = 0; c < 32; ++c) atomicAdd(y + c, xl[c] * alpha);
}

// ---------------- host-side layer driver ----------------

static void run_layer(const float* xin,         // layer input [N,128]
                      float* hbuf,              // h = lrelu(bn(xin)); later reused as accumulator
                      float* XL, float* XR, float* logitsB, unsigned* maxb, float* ssum,
                      float* stats, float* ab, const float* ea, const int* srcI,
                      const int* dstI, const float* bng, const float* bnb, const float* Wl,
                      const float* bl, const float* Wr, const float* br, const float* We,
                      const float* att, const float* bias,
                      float* final_out,         // lrelu(acc + bias) target (may alias hbuf)
                      int N, int E, hipStream_t stream) {
    const long totalNode = (long)N * 128;
    const int totalEdge = 4 * E;
    const int TB = 256;

    // BatchNorm statistics + fused apply
    zero_f32<<<1, 256, 0, stream>>>(stats, 256, 0.f);
    col_stats<<<512, 128, 0, stream>>>(xin, stats, N);
    bn_finalize<<<1, 128, 0, stream>>>(stats, bng, bnb, ab, N);
    bn_apply_lrelu<<<(int)((totalNode + TB - 1) / TB), TB, 0, stream>>>(xin, ab, hbuf,
                                                                        totalNode);

    // node transforms via fp32 WMMA (fused xl/xr)
    gemm_dual_wmma<<<dim3(N / 16, 8), 32, 0, stream>>>(hbuf, Wl, bl, Wr, br, XL, XR);

    // segment softmax over destination neighborhoods
    zero_u32<<<1024, 256, 0, stream>>>(maxb, (long)N * 4, 0u);
    edge_logits_max<<<(totalEdge + TB - 1) / TB, TB, 0, stream>>>(XL, XR, ea, srcI, dstI, We,
                                                                  att, logitsB, maxb,
                                                                  totalEdge);
    zero_f32<<<1024, 256, 0, stream>>>(ssum, (long)N * 4, 0.f);
    edge_softmax_sum<<<(totalEdge + TB - 1) / TB, TB, 0, stream>>>(logitsB, maxb, dstI, ssum,
                                                                   totalEdge);

    // weighted aggregation (hbuf is dead after the GEMM -> reuse as accumulator)
    zero_f32<<<1024, 256, 0, stream>>>(hbuf, totalNode, 0.f);
    edge_aggregate<<<(totalEdge + TB - 1) / TB, TB, 0, stream>>>(logitsB, ssum, XL, srcI,
                                                                 dstI, hbuf, totalEdge);

    add_bias_lrelu<<<(int)((totalNode + TB - 1) / TB), TB, 0, stream>>>(hbuf, bias, final_out,
                                                                        totalNode);
}

extern "C" void kernel_launch(void* const* d_in, const int* in_sizes, int n_in, void* d_out,
                              int out_size, void* d_ws, size_t ws_size, hipStream_t stream) {
    const float* x     = (const float*)d_in[0];
    const float* ea    = (const float*)d_in[1];
    const int*   eidx  = (const int*)d_in[2];
    const float* bn1g  = (const float*)d_in[3];
    const float* bn1b  = (const float*)d_in[4];
    const float* Wl1   = (const float*)d_in[5];
    const float* bl1   = (const float*)d_in[6];
    const float* Wr1   = (const float*)d_in[7];
    const float* br1   = (const float*)d_in[8];
    const float* We1   = (const float*)d_in[9];
    const float* att1  = (const float*)d_in[10];
    const float* bias1 = (const float*)d_in[11];
    const float* bn2g  = (const float*)d_in[12];
    const float* bn2b  = (const float*)d_in[13];
    const float* Wl2   = (const float*)d_in[14];
    const float* bl2   = (const float*)d_in[15];
    const float* Wr2   = (const float*)d_in[16];
    const float* br2   = (const float*)d_in[17];
    const float* We2   = (const float*)d_in[18];
    const float* att2  = (const float*)d_in[19];
    const float* bias2 = (const float*)d_in[20];

    const int N = in_sizes[0] / 128;
    const int E = in_sizes[1] / 16;
    const int* srcI = eidx;
    const int* dstI = eidx + E;

    char* wsp = (char*)d_ws;
    auto alloc = [&](size_t bytes) -> char* {
        char* p = wsp;
        wsp += (bytes + 255) & ~(size_t)255;
        return p;
    };
    float*    bufH    = (float*)alloc((size_t)N * 128 * sizeof(float));
    float*    bufXL   = (float*)alloc((size_t)N * 128 * sizeof(float));
    float*    bufXR   = (float*)alloc((size_t)N * 128 * sizeof(float));
    float*    logitsB = (float*)alloc((size_t)E * 4 * sizeof(float));
    unsigned* maxb    = (unsigned*)alloc((size_t)N * 4 * sizeof(unsigned));
    float*    ssum    = (float*)alloc((size_t)N * 4 * sizeof(float));
    float*    stats   = (float*)alloc(256 * sizeof(float));
    float*    ab      = (float*)alloc(256 * sizeof(float));

    // Layer 1: x -> y1 (stored in bufH)
    run_layer(x, bufH, bufXL, bufXR, logitsB, maxb, ssum, stats, ab, ea, srcI, dstI, bn1g,
              bn1b, Wl1, bl1, Wr1, br1, We1, att1, bias1, bufH, N, E, stream);

    // Layer 2: y1 -> d_out (bn_apply is in-place safe; bufH reused as accumulator)
    run_layer(bufH, bufH, bufXL, bufXR, logitsB, maxb, ssum, stats, ab, ea, srcI, dstI, bn2g,
              bn2b, Wl2, bl2, Wr2, br2, We2, att2, bias2, (float*)d_out, N, E, stream);
}